// ScaledDotProductAttentionWithSentenceNorm_6262062318239
// MI455X (gfx1250) — compile-verified
//
#include <hip/hip_runtime.h>

typedef float v2f __attribute__((ext_vector_type(2)));
typedef float v8f __attribute__((ext_vector_type(8)));

#define B_   4
#define H_   8
#define LQ   512
#define NB_  32
#define NT_  64
#define DP   64
#define DM   512
#define KTOT (NB_ * NT_)  /* 2048 */

static __device__ __forceinline__ v8f wmma4(v2f a, v2f b, v8f c) {
  // V_WMMA_F32_16X16X4_F32: D = A(16x4) * B(4x16) + C(16x16), fp32 throughout
  return __builtin_amdgcn_wmma_f32_16x16x4_f32(false, a, false, b, (short)0, c,
                                               false, false);
}

// ---------------------------------------------------------------------------
// Kernel A: S = (Q K^T)/8 + bias, softmax over NT, * attn_s  -> attn_w
// One wave handles a 16(q) x 64(t) tile of one (b, n, h).
// ---------------------------------------------------------------------------
__global__ __launch_bounds__(128) void attn_softmax_kernel(
    const float* __restrict__ q, const float* __restrict__ k,
    const float* __restrict__ bias, const float* __restrict__ attn_s,
    float* __restrict__ attnw /* [B,H,LQ,KTOT] */) {
  const int wave = blockIdx.x * 4 + (threadIdx.x >> 5);
  const int lane = threadIdx.x & 31;
  const int hl = lane >> 4;   // half of wave
  const int lq = lane & 15;   // lane within half

  const int qt = wave & 31;
  const int h  = (wave >> 5) & 7;
  const int n  = (wave >> 8) & 31;
  const int b  = wave >> 13;

  const float* Qb = q + (size_t)((b * H_ + h) * LQ + qt * 16) * DP;
  const float* Kb = k + (size_t)((b * NB_ + n) * H_ + h) * (NT_ * DP);
  const float* Bb = bias + (size_t)((b * NB_ + n) * H_ + h) * (LQ * NT_) +
                    (size_t)(qt * 16) * NT_;

  // Init accumulators with bias (C layout: VGPR r -> row m=r+8*hl, col=lq)
  v8f c0, c1, c2, c3;
#pragma unroll
  for (int r = 0; r < 8; ++r) {
    const int m = r + 8 * hl;
    c0[r] = Bb[(size_t)m * NT_ + 0 + lq];
    c1[r] = Bb[(size_t)m * NT_ + 16 + lq];
    c2[r] = Bb[(size_t)m * NT_ + 32 + lq];
    c3[r] = Bb[(size_t)m * NT_ + 48 + lq];
  }

  const float scale = 0.125f;  // 1/sqrt(64)
#pragma unroll 4
  for (int kk = 0; kk < 16; ++kk) {
    const int k0 = kk * 4 + 2 * hl;
    v2f a = *(const v2f*)(Qb + (size_t)lq * DP + k0);
    a.x *= scale;
    a.y *= scale;
    const v2f b0 = *(const v2f*)(Kb + (size_t)(0 + lq) * DP + k0);
    const v2f b1 = *(const v2f*)(Kb + (size_t)(16 + lq) * DP + k0);
    const v2f b2 = *(const v2f*)(Kb + (size_t)(32 + lq) * DP + k0);
    const v2f b3 = *(const v2f*)(Kb + (size_t)(48 + lq) * DP + k0);
    c0 = wmma4(a, b0, c0);
    c1 = wmma4(a, b1, c1);
    c2 = wmma4(a, b2, c2);
    c3 = wmma4(a, b3, c3);
  }

  float* Ob = attnw + (size_t)((b * H_ + h) * LQ + qt * 16) * KTOT + n * NT_;
  const float* Sb = attn_s + (size_t)((b * H_ + h) * LQ + qt * 16) * NB_ + n;

#pragma unroll
  for (int r = 0; r < 8; ++r) {
    const int m = r + 8 * hl;
    const float x0 = c0[r], x1 = c1[r], x2 = c2[r], x3 = c3[r];
    float mx = fmaxf(fmaxf(x0, x1), fmaxf(x2, x3));
#pragma unroll
    for (int msk = 1; msk < 16; msk <<= 1)
      mx = fmaxf(mx, __shfl_xor(mx, msk, 32));  // stays within 16-lane halves
    const float e0 = __expf(x0 - mx), e1 = __expf(x1 - mx);
    const float e2 = __expf(x2 - mx), e3 = __expf(x3 - mx);
    float s = (e0 + e1) + (e2 + e3);
#pragma unroll
    for (int msk = 1; msk < 16; msk <<= 1) s += __shfl_xor(s, msk, 32);
    const float w = Sb[(size_t)m * NB_] / s;
    Ob[(size_t)m * KTOT + 0 + lq] = e0 * w;
    Ob[(size_t)m * KTOT + 16 + lq] = e1 * w;
    Ob[(size_t)m * KTOT + 32 + lq] = e2 * w;
    Ob[(size_t)m * KTOT + 48 + lq] = e3 * w;
  }
}

// ---------------------------------------------------------------------------
// Kernel B: out1 = attn_w[512x2048] * Vw[2048x64] per (b,h); X -> d_ws in
// [b*LQ + q, h*64 + d] layout (ready for the FC).
// ---------------------------------------------------------------------------
__global__ __launch_bounds__(128) void attn_v_kernel(
    const float* __restrict__ attnw, const float* __restrict__ v,
    float* __restrict__ X /* [B*LQ, DM] */) {
  const int lane = threadIdx.x & 31;
  const int hl = lane >> 4;
  const int lq = lane & 15;

  const int bh = blockIdx.x >> 3;                              // 0..31
  const int qt = ((blockIdx.x & 7) << 2) + (threadIdx.x >> 5); // 0..31
  const int b = bh >> 3, h = bh & 7;

  const float* Ab = attnw + (size_t)((b * H_ + h) * LQ + qt * 16) * KTOT;

  v8f c0 = {0}, c1 = {0}, c2 = {0}, c3 = {0};
#pragma unroll 2
  for (int k0 = 0; k0 < KTOT; k0 += 4) {
    const int nblk = k0 >> 6;
    const int t0 = (k0 & 63) + 2 * hl;
    const float* Vk =
        v + (size_t)((b * NB_ + nblk) * H_ + h) * (NT_ * DP) + (size_t)t0 * DP;
    const v2f a = *(const v2f*)(Ab + (size_t)lq * KTOT + k0 + 2 * hl);
    const v2f b0 = {Vk[0 + lq], Vk[DP + 0 + lq]};
    const v2f b1 = {Vk[16 + lq], Vk[DP + 16 + lq]};
    const v2f b2 = {Vk[32 + lq], Vk[DP + 32 + lq]};
    const v2f b3 = {Vk[48 + lq], Vk[DP + 48 + lq]};
    c0 = wmma4(a, b0, c0);
    c1 = wmma4(a, b1, c1);
    c2 = wmma4(a, b2, c2);
    c3 = wmma4(a, b3, c3);
  }

  float* Xb = X + (size_t)(b * LQ + qt * 16) * DM + h * DP;
#pragma unroll
  for (int r = 0; r < 8; ++r) {
    const int m = r + 8 * hl;
    Xb[(size_t)m * DM + 0 + lq] = c0[r];
    Xb[(size_t)m * DM + 16 + lq] = c1[r];
    Xb[(size_t)m * DM + 32 + lq] = c2[r];
    Xb[(size_t)m * DM + 48 + lq] = c3[r];
  }
}

// ---------------------------------------------------------------------------
// Kernel C: out = X[2048x512] * W_fc^T + b_fc. B-frag rows are W_fc rows, so
// the contraction dim is contiguous -> float2 loads per lane.
// ---------------------------------------------------------------------------
__global__ __launch_bounds__(128) void fc_kernel(
    const float* __restrict__ X, const float* __restrict__ W,
    const float* __restrict__ bfc, float* __restrict__ out) {
  const int wave = blockIdx.x * 4 + (threadIdx.x >> 5);
  const int lane = threadIdx.x & 31;
  const int hl = lane >> 4;
  const int lq = lane & 15;

  const int ngrp = wave & 7;   // 64-col group
  const int mt = wave >> 3;    // 0..127 row tiles
  const int j0 = ngrp * 64;

  const float* Xb = X + (size_t)(mt * 16) * DM;

  v8f c0 = {0}, c1 = {0}, c2 = {0}, c3 = {0};
#pragma unroll 4
  for (int kc = 0; kc < DM; kc += 4) {
    const int k0 = kc + 2 * hl;
    const v2f a = *(const v2f*)(Xb + (size_t)lq * DM + k0);
    const v2f b0 = *(const v2f*)(W + (size_t)(j0 + 0 + lq) * DM + k0);
    const v2f b1 = *(const v2f*)(W + (size_t)(j0 + 16 + lq) * DM + k0);
    const v2f b2 = *(const v2f*)(W + (size_t)(j0 + 32 + lq) * DM + k0);
    const v2f b3 = *(const v2f*)(W + (size_t)(j0 + 48 + lq) * DM + k0);
    c0 = wmma4(a, b0, c0);
    c1 = wmma4(a, b1, c1);
    c2 = wmma4(a, b2, c2);
    c3 = wmma4(a, b3, c3);
  }

  const float bb0 = bfc[j0 + 0 + lq], bb1 = bfc[j0 + 16 + lq];
  const float bb2 = bfc[j0 + 32 + lq], bb3 = bfc[j0 + 48 + lq];
  float* Ob = out + (size_t)(mt * 16) * DM + j0;
#pragma unroll
  for (int r = 0; r < 8; ++r) {
    const int m = r + 8 * hl;
    Ob[(size_t)m * DM + 0 + lq] = c0[r] + bb0;
    Ob[(size_t)m * DM + 16 + lq] = c1[r] + bb1;
    Ob[(size_t)m * DM + 32 + lq] = c2[r] + bb2;
    Ob[(size_t)m * DM + 48 + lq] = c3[r] + bb3;
  }
}

extern "C" void kernel_launch(void* const* d_in, const int* in_sizes, int n_in,
                              void* d_out, int out_size, void* d_ws,
                              size_t ws_size, hipStream_t stream) {
  const float* q = (const float*)d_in[0];
  const float* k = (const float*)d_in[1];
  const float* v = (const float*)d_in[2];
  const float* attn_s = (const float*)d_in[3];
  const float* bias = (const float*)d_in[4];
  const float* W_fc = (const float*)d_in[5];
  const float* b_fc = (const float*)d_in[6];

  float* out = (float*)d_out;                         // [B, LQ, DM]
  float* attnw = out + (size_t)B_ * LQ * DM;          // [B, H, LQ, KTOT]
  float* X = (float*)d_ws;                            // [B*LQ, DM] scratch

  // A: B*NB*H*(LQ/16) = 32768 waves -> 8192 blocks of 4 waves
  attn_softmax_kernel<<<8192, 128, 0, stream>>>(q, k, bias, attn_s, attnw);
  // B: B*H*(LQ/16) = 1024 waves -> 256 blocks (4 q-tiles share (b,h) per block)
  attn_v_kernel<<<256, 128, 0, stream>>>(attnw, v, X);
  // C: 128 m-tiles * 8 n-groups = 1024 waves -> 256 blocks
  fc_kernel<<<256, 128, 0, stream>>>(X, W_fc, b_fc, out);
}